// QSScan_24524263260383
// MI455X (gfx1250) — compile-verified
//
#include <hip/hip_runtime.h>

// ---------------------------------------------------------------------------
// Quantized selective scan (Mamba) for MI455X / gfx1250.
//   sizes: B_SZ=4, DIM=4096, DSTATE=16, L=2048
// Memory-bound (~540MB => ~23us @ 23.3TB/s) + ~640M v_exp_f32.
// One lane per (b,d) channel, 16 states in VGPRs; B/C staged to LDS via the
// Tensor Data Mover (double buffered, TENSORcnt), u/dt/z streamed per-lane
// with int4 loads + global_prefetch_b8 hints.
// ---------------------------------------------------------------------------

#define B_SZ   4
#define DIM    4096
#define DSTATE 16
#define SEQL   2048
#define TT     128              // timesteps per LDS tile
#define NT     (SEQL / TT)      // 16 tiles
#define BLK    64               // 2 waves per block

// ---- fast scalar math (native gfx1250 TRANS ops) ---------------------------
__device__ __forceinline__ float fexp(float x) {        // e^x via v_exp_f32
    return __builtin_amdgcn_exp2f(x * 1.4426950408889634f);
}
__device__ __forceinline__ float fln(float x) {         // ln via v_log_f32
    return __builtin_amdgcn_logf(x) * 0.6931471805599453f;
}
__device__ __forceinline__ float frcp(float x) {
    return __builtin_amdgcn_rcpf(x);
}

// ---- Tensor Data Mover -----------------------------------------------------
// The TDM builtin only exists in the DEVICE compilation pass; the host pass
// parses this TU too, so all builtin references are fenced behind
// __HIP_DEVICE_COMPILE__.  Absence on the device pass is a hard error so a
// successful compile proves tensor_load_to_lds is in the binary.
#if defined(__HIP_DEVICE_COMPILE__)
#  if !defined(__has_builtin) || !__has_builtin(__builtin_amdgcn_tensor_load_to_lds)
#    error "gfx1250 device pass lacks __builtin_amdgcn_tensor_load_to_lds"
#  endif
#endif

typedef unsigned int u32x4 __attribute__((ext_vector_type(4)));
typedef int          i32x8 __attribute__((ext_vector_type(8)));
typedef int          i32x4 __attribute__((ext_vector_type(4)));

// 1-D TDM copy: nelem 4-byte elements from global -> LDS (D# per ISA §8).
__device__ __forceinline__ void tdm_load_1d(unsigned int lds_off,
                                            const void* gsrc,
                                            unsigned int nelem) {
#if defined(__HIP_DEVICE_COMPILE__)
    unsigned long long ga = (unsigned long long)(uintptr_t)gsrc;
    u32x4 g0;
    g0[0] = 1u;                                            // count=1, user D#
    g0[1] = lds_off;                                       // lds_addr
    g0[2] = (unsigned int)ga;                              // global_addr[31:0]
    g0[3] = (unsigned int)((ga >> 32) & 0x01FFFFFFull)     // global_addr[56:32]
          | (2u << 30);                                    // type=2 ("image")
    i32x8 g1;
    g1[0] = (int)(2u << 16);                               // data_size=4B
    g1[1] = (int)((nelem & 0xFFFFu) << 16);                // tensor_dim0[15:0]
    g1[2] = (int)(nelem >> 16);                            // tensor_dim0[31:16]
    g1[3] = (int)((nelem & 0xFFFFu) << 16);                // tile_dim0 = nelem
    g1[4] = 0;                                             // tile_dim1/2 = 0
    g1[5] = (int)nelem;                                    // dim0_stride (1-D)
    g1[6] = 0;
    g1[7] = 0;
    i32x4 z4 = (i32x4)0;
#  if defined(__clang_major__) && (__clang_major__ >= 23)
    i32x8 z8 = (i32x8)0;
    __builtin_amdgcn_tensor_load_to_lds(g0, g1, z4, z4, z8, 0);
#  else
    __builtin_amdgcn_tensor_load_to_lds(g0, g1, z4, z4, 0);
#  endif
#else
    (void)lds_off; (void)gsrc; (void)nelem;   // host pass: never executed
#endif
}

__device__ __forceinline__ void tdm_wait_le2() {
#if defined(__HIP_DEVICE_COMPILE__)
    __builtin_amdgcn_s_wait_tensorcnt(2);
#endif
}
__device__ __forceinline__ void tdm_wait_le0() {
#if defined(__HIP_DEVICE_COMPILE__)
    __builtin_amdgcn_s_wait_tensorcnt(0);
#endif
}

// ---------------------------------------------------------------------------
// Main scan kernel: grid (DIM/BLK, B_SZ), block BLK. Lane == channel d.
// (Defined FIRST so the device-asm snippet shows this kernel.)
// ---------------------------------------------------------------------------
__global__ __launch_bounds__(BLK)
void qss_scan(const int* __restrict__ u,   const int* __restrict__ dt,
              const int* __restrict__ z,   const int* __restrict__ A_log,
              const int* __restrict__ D_q, const int* __restrict__ dt_bias,
              const float* __restrict__ Btf, const float* __restrict__ Ctf,
              const float* __restrict__ u_sp,  const float* __restrict__ dt_sp,
              const float* __restrict__ z_sp,  const float* __restrict__ A_sp,
              const float* __restrict__ D_sp,  const float* __restrict__ dtb_sp,
              float* __restrict__ out) {
    const int b = blockIdx.y;
    const int d = blockIdx.x * BLK + threadIdx.x;

    const float u_scale   = *u_sp;
    const float dt_scale  = *dt_sp;
    const float z_scale   = *z_sp;
    const float A_scale   = *A_sp;
    const float D_scale   = *D_sp;
    const float dtb_scale = *dtb_sp;

    __shared__ __align__(16) float sB[2][TT * DSTATE];   // 2 x 8KB
    __shared__ __align__(16) float sC[2][TT * DSTATE];   // 2 x 8KB

    // per-channel constants
    float Af[DSTATE];
#pragma unroll
    for (int n = 0; n < DSTATE; ++n)
        Af[n] = -fexp((float)A_log[d * DSTATE + n] * A_scale);
    const float Df  = (float)D_q[d] * D_scale;
    const float dtb = (float)dt_bias[d] * dtb_scale;

    float h[DSTATE];
#pragma unroll
    for (int n = 0; n < DSTATE; ++n) h[n] = 0.0f;

    const size_t chan = (size_t)b * DIM + d;
    const int4*  up4 = (const int4*)(u  + chan * SEQL);
    const int4*  dp4 = (const int4*)(dt + chan * SEQL);
    const int4*  zp4 = (const int4*)(z  + chan * SEQL);
    float4*      op4 = (float4*)(out + chan * SEQL);

    const float* Bt_b = Btf + (size_t)b * SEQL * DSTATE;
    const float* Ct_b = Ctf + (size_t)b * SEQL * DSTATE;

    if (threadIdx.x < 32) {   // wave 0 drives the TDM pipeline
        tdm_load_1d((unsigned int)(uintptr_t)&sB[0][0], Bt_b, TT * DSTATE);
        tdm_load_1d((unsigned int)(uintptr_t)&sC[0][0], Ct_b, TT * DSTATE);
    }

    for (int tile = 0; tile < NT; ++tile) {
        const int cur = tile & 1;
        __syncthreads();  // all waves done reading buffer cur^1 (tile-1)
        if (threadIdx.x < 32) {
            if (tile + 1 < NT) {
                const size_t nxt = (size_t)(tile + 1) * TT * DSTATE;
                tdm_load_1d((unsigned int)(uintptr_t)&sB[cur ^ 1][0],
                            Bt_b + nxt, TT * DSTATE);
                tdm_load_1d((unsigned int)(uintptr_t)&sC[cur ^ 1][0],
                            Ct_b + nxt, TT * DSTATE);
                tdm_wait_le2();   // the two loads for tile 'tile' have landed
            } else {
                tdm_wait_le0();
            }
        }
        __syncthreads();  // buffer cur ready for all waves

        // prefetch next tile of this lane's u/dt/z streams (global_prefetch_b8)
        if (tile + 1 < NT) {
            const char* pu = (const char*)up4 + (size_t)(tile + 1) * TT * 4;
            const char* pd = (const char*)dp4 + (size_t)(tile + 1) * TT * 4;
            const char* pz = (const char*)zp4 + (size_t)(tile + 1) * TT * 4;
#pragma unroll
            for (int j = 0; j < TT * 4; j += 128) {
                __builtin_prefetch(pu + j, 0, 3);
                __builtin_prefetch(pd + j, 0, 3);
                __builtin_prefetch(pz + j, 0, 3);
            }
        }

        const int base4 = tile * (TT / 4);
        for (int q = 0; q < TT / 4; ++q) {
            int4 u4 = up4[base4 + q];
            int4 t4 = dp4[base4 + q];
            int4 z4 = zp4[base4 + q];
            int uu[4] = {u4.x, u4.y, u4.z, u4.w};
            int td[4] = {t4.x, t4.y, t4.z, t4.w};
            int zv[4] = {z4.x, z4.y, z4.z, z4.w};
            float yo[4];
#pragma unroll
            for (int k = 0; k < 4; ++k) {
                const int t = q * 4 + k;
                // softplus(dt*scale + bias)
                float x   = (float)td[k] * dt_scale + dtb;
                float ex  = fexp(x);
                float dtf = (x > 16.0f) ? x : fln(1.0f + ex);
                float uf  = (float)uu[k] * u_scale;
                float du  = dtf * uf;

                float Bl[DSTATE], Cl[DSTATE];
#pragma unroll
                for (int v = 0; v < 4; ++v) {
                    *(float4*)&Bl[4 * v] =
                        *(const float4*)&sB[cur][t * DSTATE + 4 * v];
                    *(float4*)&Cl[4 * v] =
                        *(const float4*)&sC[cur][t * DSTATE + 4 * v];
                }

                float acc = 0.0f;
#pragma unroll
                for (int n = 0; n < DSTATE; ++n) {
                    float w = fexp(dtf * Af[n]);            // v_exp_f32
                    h[n] = fmaf(w, h[n], du * Bl[n]);
                    acc  = fmaf(h[n], Cl[n], acc);
                }

                float y  = fmaf(uf, Df, acc);               // skip connection
                float zf = (float)zv[k] * z_scale;          // SiLU gate
                float sg = frcp(1.0f + fexp(-zf));
                yo[k] = y * (zf * sg);
            }
            float4 o;
            o.x = yo[0]; o.y = yo[1]; o.z = yo[2]; o.w = yo[3];
            op4[base4 + q] = o;
        }
    }
}

// ---------------------------------------------------------------------------
// Prepass: dequantize + transpose B,C: (b, n, L) int32 -> (b, L, n) float
// ---------------------------------------------------------------------------
__global__ void qss_dq_transpose(const int* __restrict__ B,
                                 const int* __restrict__ C,
                                 const float* __restrict__ Bs,
                                 const float* __restrict__ Cs,
                                 float* __restrict__ Bt,
                                 float* __restrict__ Ct) {
    const int total = B_SZ * DSTATE * SEQL;
    int idx = blockIdx.x * blockDim.x + threadIdx.x;
    if (idx >= total) return;
    const float bs = *Bs, cs = *Cs;
    int t = idx % SEQL;
    int n = (idx / SEQL) % DSTATE;
    int b = idx / (SEQL * DSTATE);
    size_t o = ((size_t)b * SEQL + t) * DSTATE + n;
    Bt[o] = (float)B[idx] * bs;
    Ct[o] = (float)C[idx] * cs;
}

// ---------------------------------------------------------------------------
extern "C" void kernel_launch(void* const* d_in, const int* in_sizes, int n_in,
                              void* d_out, int out_size, void* d_ws, size_t ws_size,
                              hipStream_t stream) {
    const int*   u        = (const int*)d_in[0];
    const int*   dt       = (const int*)d_in[1];
    const int*   Bq       = (const int*)d_in[2];
    const int*   Cq       = (const int*)d_in[3];
    const int*   z        = (const int*)d_in[4];
    const int*   A_log    = (const int*)d_in[5];
    const int*   D_q      = (const int*)d_in[6];
    const int*   dt_bias  = (const int*)d_in[7];
    const float* u_scale  = (const float*)d_in[8];
    const float* dt_scale = (const float*)d_in[9];
    const float* B_scale  = (const float*)d_in[10];
    const float* C_scale  = (const float*)d_in[11];
    const float* z_scale  = (const float*)d_in[12];
    const float* A_scale  = (const float*)d_in[13];
    const float* D_scale  = (const float*)d_in[14];
    const float* dtb_scale= (const float*)d_in[15];
    float* out = (float*)d_out;

    // workspace: dequantized transposed B,C as (b, L, n) float
    float* Btf = (float*)d_ws;
    float* Ctf = Btf + (size_t)B_SZ * SEQL * DSTATE;

    const int total = B_SZ * DSTATE * SEQL;
    qss_dq_transpose<<<(total + 255) / 256, 256, 0, stream>>>(
        Bq, Cq, B_scale, C_scale, Btf, Ctf);

    dim3 grid(DIM / BLK, B_SZ);
    qss_scan<<<grid, BLK, 0, stream>>>(
        u, dt, z, A_log, D_q, dt_bias, Btf, Ctf,
        u_scale, dt_scale, z_scale, A_scale, D_scale, dtb_scale, out);
}